// CARAFE_68341519614297
// MI455X (gfx1250) — compile-verified
//
#include <hip/hip_runtime.h>

// CARAFE upsample (KS=5, GS=1, SF=2) via V_WMMA_F32_16X16X4_F32 on gfx1250.
//
// One wave = one (n, 2x2 input-pixel) tile -> 4x4 output pixels.
//   M = 16 channels, N = 16 output pixels (4x4), K = 36 (6x6 feature window,
//   9 WMMA steps of K=4).  B (masks, 25/36 dense) is built once per wave with
//   BOTH the tap-validity zeros and the image-boundary padding zeros folded
//   in; A (features) then loads pairs with clamped addresses and no fixup.
//   A pairs (k,k+1) share a row and start at even x (wi even), so each pair
//   is fully in- or out-of-bounds and 8B aligned -> one global_load_b64.

typedef __attribute__((ext_vector_type(2))) float v2f;
typedef __attribute__((ext_vector_type(8))) float v8f;

constexpr int CH  = 256;
constexpr int HI  = 64, WI = 64;
constexpr int OH  = 128, OW = 128;
constexpr int KS  = 5,  KK = 25;
constexpr int HW  = HI * WI;     // 4096
constexpr int OHW = OH * OW;     // 16384

__global__ __launch_bounds__(256) void carafe_wmma_f32(
    const float* __restrict__ feat,   // [8][256][64][64]
    const float* __restrict__ mask,   // [8][25][128][128]
    float* __restrict__ out)          // [8][256][128][128]
{
    const int lane      = threadIdx.x & 31;
    const int waveInBlk = threadIdx.x >> 5;
    const int wave      = blockIdx.x * 8 + waveInBlk;   // 0..8191

    // tile decomposition: 8 batches x 32x32 tiles of 2x2 input pixels
    const int n  = wave >> 10;
    const int r  = wave & 1023;
    const int th = r >> 5;
    const int tw = r & 31;
    const int hi  = th * 2;     // input tile origin (rows hi..hi+1)
    const int wi  = tw * 2;     // even -> A pairs never straddle the border
    const int oh0 = th * 4;
    const int ow0 = tw * 4;

    const int half = lane >> 4;   // upper half-wave holds K rows +2 per step
    const int l    = lane & 15;   // N column (B/D) and M row (A)

    // This lane's output pixel (N column l): p = ph*4+pw
    const int ph = l >> 2, pw = l & 3;
    const int oh = oh0 + ph, ow = ow0 + pw;
    const int dyb = ph >> 1;      // owning input pixel inside 2x2 tile
    const int dxb = pw >> 1;

    // ---------------- B: masks with tap + padding zeros folded in ---------
    // B[k=(fy*6+fx), p] = mask[n, dy*5+dx, oh, ow], zeroed unless the tap
    // (dy,dx)=(fy-dyb,fx-dxb) is in [0,5)^2 AND feature (hi-2+fy, wi-2+fx)
    // is inside the image (this replaces per-tile zeroing of A).
    v2f B[9];
    const float* mb = mask + ((size_t)n * KK * OH + (size_t)oh) * OW + ow;
#pragma unroll
    for (int s = 0; s < 9; ++s) {
#pragma unroll
        for (int e = 0; e < 2; ++e) {
            const int k  = 4 * s + 2 * half + e;
            const int fy = k / 6, fx = k % 6;
            const int dy = fy - dyb, dx = fx - dxb;
            const int gr = hi - 2 + fy, gc = wi - 2 + fx;
            const bool tap = ((unsigned)dy < KS) && ((unsigned)dx < KS);
            const bool img = ((unsigned)gr < HI) && ((unsigned)gc < WI);
            const int  kc  = tap ? (dy * KS + dx) : 0;   // clamped index
            B[s][e] = mb[(size_t)kc * OHW] * ((tap && img) ? 1.0f : 0.0f);
        }
    }

    // -------- A pair offsets: 32-bit per-lane byte offsets, fixed ---------
    unsigned aoff[9];
#pragma unroll
    for (int s = 0; s < 9; ++s) {
        const int k0 = 4 * s + 2 * half;     // pair start, fx even
        const int fy = k0 / 6, fx = k0 % 6;
        const int gr = hi - 2 + fy, gc = wi - 2 + fx;
        const bool img = ((unsigned)gr < HI) && ((unsigned)gc < WI);
        // OOB pairs read offset 0 (garbage); zeroed via B.
        aoff[s] = (unsigned)((l * HW + (img ? gr * WI + gc : 0)) * 4);
    }

    // uniform scalar bases, advanced per channel tile (SALU); per-lane
    // 32-bit offsets stay fixed -> GVS addressing, no per-iter 64-bit VALU.
    const char* fbase = (const char*)(feat + (size_t)n * CH * HW);
    char*       obase = (char*)(out + (size_t)n * CH * OHW);
    const unsigned soff =
        (unsigned)(((8 * half) * OHW + oh * OW + ow) * 4);

    // ---------------- channel-tile loop: 9 b64 loads, 9 WMMA, 8 stores ----
#pragma unroll 2
    for (int ct = 0; ct < 16; ++ct) {
        v2f A[9];
#pragma unroll
        for (int s = 0; s < 9; ++s)
            A[s] = *(const v2f*)(fbase + aoff[s]);

        v8f acc = {};
#pragma unroll
        for (int s = 0; s < 9; ++s) {
            // (neg_a, A, neg_b, B, c_mod, C, reuse_a, reuse_b)
            acc = __builtin_amdgcn_wmma_f32_16x16x4_f32(
                false, A[s], false, B[s], (short)0, acc, false, false);
        }

        // D: VGPR j -> channel ct*16 + j + 8*half, column = this lane's pixel
#pragma unroll
        for (int j = 0; j < 8; ++j)
            *(float*)(obase + (soff + (unsigned)(j * OHW * 4))) = acc[j];

        fbase += 16 * HW * 4;    // 0x40000: next 16-channel feature slab
        obase += 16 * OHW * 4;   // 0x100000: next 16-channel output slab
    }
}

extern "C" void kernel_launch(void* const* d_in, const int* in_sizes, int n_in,
                              void* d_out, int out_size, void* d_ws, size_t ws_size,
                              hipStream_t stream) {
    const float* feat = (const float*)d_in[0];   // (8,256,64,64) fp32
    const float* mask = (const float*)d_in[1];   // (8,25,128,128) fp32
    float*       out  = (float*)d_out;           // (8,256,128,128) fp32

    // 8192 waves = 8 batches * 32*32 tiles; 8 waves per 256-thread block.
    dim3 grid(1024), block(256);
    hipLaunchKernelGGL(carafe_wmma_f32, grid, block, 0, stream,
                       feat, mask, out);
}